// LessSimplifiedGNN_61272003444819
// MI455X (gfx1250) — compile-verified
//
#include <hip/hip_runtime.h>
#include <hip/hip_bf16.h>
#include <cstdint>
#include <cstddef>

typedef __attribute__((ext_vector_type(2))) float v2f;
typedef __attribute__((ext_vector_type(8))) float v8f;

// ---------------------------------------------------------------------------
// Degree / normalization:  deg[v] = 1 (self loop) + #incoming edges;
// dinv[v] = rsqrt(deg[v])   (deg >= 1 always, so no zero guard needed)
// ---------------------------------------------------------------------------
__global__ void k_deg_init(float* __restrict__ deg, int n) {
    int i = blockIdx.x * blockDim.x + threadIdx.x;
    if (i < n) deg[i] = 1.0f;
}

__global__ void k_deg_count(const long long* __restrict__ ei, float* __restrict__ deg,
                            long long E) {
    long long e = (long long)blockIdx.x * blockDim.x + threadIdx.x;
    if (e < E) atomicAdd(&deg[(int)ei[E + e]], 1.0f);   // col = target
}

__global__ void k_deg_rsqrt(float* __restrict__ deg, int n) {
    int i = blockIdx.x * blockDim.x + threadIdx.x;
    if (i < n) deg[i] = rsqrtf(deg[i]);
}

// ---------------------------------------------------------------------------
// GEMM1: h1[N,12] = x[N,128] @ W1[128,12] via V_WMMA_F32_16X16X4_F32.
// One wave per 16-row tile; 32 WMMAs walk K=128.
//
// W1 is staged in LDS, zero-padded to 16 columns and pre-swizzled so each
// lane's B fragment is a contiguous v2f:
//   sW[j*32 + n*2 + p] = (n < 12) ? W1[(2j+p)*12 + n] : 0,  j = K-pair index.
// Lane layout (ISA 32-bit 16x4 A / 4x16 B):
//   m = lane&15, hi = lane>>4; A holds K = k+2*hi+{0,1}; B mirrors with
//   n = lane&15; C/D VGPR i -> row i+8*hi, col lane&15.
// ---------------------------------------------------------------------------
__global__ void k_gemm1_wmma(const float* __restrict__ x, const float* __restrict__ W1,
                             float* __restrict__ h1, int ntiles, int N) {
    __shared__ float sW[64 * 32];   // 64 K-pairs x 16 cols x 2 = 8 KB
    for (int i = threadIdx.x; i < 64 * 32; i += blockDim.x) {
        const int j = i >> 5, rem = i & 31, n = rem >> 1, p = rem & 1;
        sW[i] = (n < 12) ? W1[(2 * j + p) * 12 + n] : 0.0f;
    }
    __syncthreads();

    const int lane = threadIdx.x & 31;
    const int wave = threadIdx.x >> 5;
    const int tile = blockIdx.x * (blockDim.x >> 5) + wave;
    if (tile >= ntiles) return;                 // wave-uniform exit

    const int m  = lane & 15;
    const int hi = lane >> 4;
    const int n  = m;
    int row = tile * 16 + m;
    if (row >= N) row = N - 1;                  // clamp (masked on store)
    const float* __restrict__ xr = x + (size_t)row * 128 + 2 * hi;
    const float* __restrict__ bp = sW + n * 2 + hi * 32;   // + (k/2)*32 per step

    v8f acc = {0.f, 0.f, 0.f, 0.f, 0.f, 0.f, 0.f, 0.f};

#pragma unroll
    for (int k = 0; k < 128; k += 4) {
        v2f a = *(const v2f*)(xr + k);                 // global_load_b64
        v2f b = *(const v2f*)(bp + (k >> 1) * 32);     // ds_load_b64
        acc = __builtin_amdgcn_wmma_f32_16x16x4_f32(false, a, false, b,
                                                    (short)0, acc, false, false);
    }

    if (n < 12) {
#pragma unroll
        for (int i = 0; i < 8; ++i) {
            const int r = tile * 16 + i + 8 * hi;
            if (r < N) h1[(size_t)r * 12 + n] = acc[i];
        }
    }
}

// ---------------------------------------------------------------------------
// GEMM2: h2[N,24] = relu(agg1[N,12] + b1) @ W2[12,24]  (ReLU+bias fused into
// the A-fragment load). K=12 -> 3 K-steps; N=24 -> 2 N-tiles (cols 24..31
// zero-padded in LDS). 6 WMMAs per tile.
// ---------------------------------------------------------------------------
__global__ void k_gemm2_wmma(const float* __restrict__ agg1, const float* __restrict__ b1,
                             const float* __restrict__ W2, float* __restrict__ h2,
                             int ntiles, int N) {
    __shared__ float sW[6 * 64];   // 6 K-pairs x 32 cols x 2 = 1.5 KB
    __shared__ float sB[12];
    for (int i = threadIdx.x; i < 6 * 64; i += blockDim.x) {
        const int j = i >> 6, rem = i & 63, c = rem >> 1, p = rem & 1;
        sW[i] = (c < 24) ? W2[(2 * j + p) * 24 + c] : 0.0f;
    }
    if (threadIdx.x < 12) sB[threadIdx.x] = b1[threadIdx.x];
    __syncthreads();

    const int lane = threadIdx.x & 31;
    const int wave = threadIdx.x >> 5;
    const int tile = blockIdx.x * (blockDim.x >> 5) + wave;
    if (tile >= ntiles) return;

    const int m  = lane & 15;
    const int hi = lane >> 4;
    const int n  = m;
    int row = tile * 16 + m;
    if (row >= N) row = N - 1;
    const float* __restrict__ ar = agg1 + (size_t)row * 12 + 2 * hi;

    v8f acc0 = {0.f, 0.f, 0.f, 0.f, 0.f, 0.f, 0.f, 0.f};
    v8f acc1 = {0.f, 0.f, 0.f, 0.f, 0.f, 0.f, 0.f, 0.f};

#pragma unroll
    for (int k = 0; k < 12; k += 4) {
        const int j = (k >> 1) + hi;                   // K-pair index
        v2f av   = *(const v2f*)(ar + k);              // global_load_b64
        v2f bias = *(const v2f*)(sB + 2 * j);          // ds_load_b64
        v2f a;
        a.x = fmaxf(av.x + bias.x, 0.0f);
        a.y = fmaxf(av.y + bias.y, 0.0f);
        v2f bA = *(const v2f*)(sW + j * 64 + n * 2);          // cols 0..15
        v2f bB = *(const v2f*)(sW + j * 64 + (16 + n) * 2);   // cols 16..31
        acc0 = __builtin_amdgcn_wmma_f32_16x16x4_f32(false, a, false, bA,
                                                     (short)0, acc0, false, false);
        acc1 = __builtin_amdgcn_wmma_f32_16x16x4_f32(false, a, false, bB,
                                                     (short)0, acc1, false, false);
    }

#pragma unroll
    for (int i = 0; i < 8; ++i) {
        const int r = tile * 16 + i + 8 * hi;
        if (r < N) {
            h2[(size_t)r * 24 + n] = acc0[i];
            if (n < 8) h2[(size_t)r * 24 + 16 + n] = acc1[i];
        }
    }
}

// ---------------------------------------------------------------------------
// Self-loop pass: out[v,:] = h[v,:] * dinv[v]^2  (fully overwrites out ->
// deterministic across graph replays, no memset needed).
// ---------------------------------------------------------------------------
template <int F>
__global__ void k_selfloop(const float* __restrict__ h, const float* __restrict__ dinv,
                           float* __restrict__ out, int n) {
    int v = blockIdx.x * blockDim.x + threadIdx.x;
    if (v >= n) return;
    float d = dinv[v];
    d = d * d;
    const float4* __restrict__ src = (const float4*)(h + (size_t)v * F);
    float4* __restrict__ dst = (float4*)(out + (size_t)v * F);
#pragma unroll
    for (int g = 0; g < F / 4; ++g) {
        float4 t = src[g];
        t.x *= d; t.y *= d; t.z *= d; t.w *= d;
        dst[g] = t;
    }
}

// ---------------------------------------------------------------------------
// Edge scatter: out[col,:] += h[row,:] * dinv[row]*dinv[col], one thread per
// edge, float4 gathers + f32 atomics (agg buffers fit in the 192 MB L2).
// ---------------------------------------------------------------------------
template <int F>
__global__ void k_scatter(const long long* __restrict__ ei, const float* __restrict__ h,
                          const float* __restrict__ dinv, float* __restrict__ out,
                          long long E) {
    long long e = (long long)blockIdx.x * blockDim.x + threadIdx.x;
    if (e >= E) return;
    const int s = (int)ei[e];        // source (row)
    const int t = (int)ei[E + e];    // target (col)
    const float nrm = dinv[s] * dinv[t];
    const float4* __restrict__ hs = (const float4*)(h + (size_t)s * F);
    float* __restrict__ dst = out + (size_t)t * F;
#pragma unroll
    for (int g = 0; g < F / 4; ++g) {
        float4 v4 = hs[g];
        atomicAdd(dst + 4 * g + 0, v4.x * nrm);
        atomicAdd(dst + 4 * g + 1, v4.y * nrm);
        atomicAdd(dst + 4 * g + 2, v4.z * nrm);
        atomicAdd(dst + 4 * g + 3, v4.w * nrm);
    }
}

// ---------------------------------------------------------------------------
// Fused MLP tail: a2 = relu(agg2 + b2); h3 = relu(a2@Wf1 + bf1);
// out = h3@Wf2 + bf2.  Weights staged in LDS once per block.
// ---------------------------------------------------------------------------
__global__ void k_mlp(const float* __restrict__ agg2, const float* __restrict__ b2,
                      const float* __restrict__ Wf1, const float* __restrict__ bf1,
                      const float* __restrict__ Wf2, const float* __restrict__ bf2,
                      float* __restrict__ out, int n) {
    __shared__ float sW1[24 * 32];
    __shared__ float sW2[32 * 2];
    __shared__ float sb1[32];
    __shared__ float sb0[24];
    __shared__ float sb2[2];
    for (int i = threadIdx.x; i < 24 * 32; i += blockDim.x) sW1[i] = Wf1[i];
    for (int i = threadIdx.x; i < 64; i += blockDim.x) sW2[i] = Wf2[i];
    if (threadIdx.x < 32) sb1[threadIdx.x] = bf1[threadIdx.x];
    if (threadIdx.x < 24) sb0[threadIdx.x] = b2[threadIdx.x];
    if (threadIdx.x < 2)  sb2[threadIdx.x] = bf2[threadIdx.x];
    __syncthreads();

    const int v = blockIdx.x * blockDim.x + threadIdx.x;
    if (v >= n) return;

    float a[24];
#pragma unroll
    for (int f = 0; f < 24; ++f)
        a[f] = fmaxf(agg2[(size_t)v * 24 + f] + sb0[f], 0.0f);

    float o0 = sb2[0], o1 = sb2[1];
#pragma unroll
    for (int j = 0; j < 32; ++j) {
        float hj = sb1[j];
#pragma unroll
        for (int f = 0; f < 24; ++f) hj = fmaf(a[f], sW1[f * 32 + j], hj);
        hj = fmaxf(hj, 0.0f);
        o0 = fmaf(hj, sW2[j * 2 + 0], o0);
        o1 = fmaf(hj, sW2[j * 2 + 1], o1);
    }
    out[(size_t)v * 2 + 0] = o0;
    out[(size_t)v * 2 + 1] = o1;
}

// ---------------------------------------------------------------------------
// Launch
// ---------------------------------------------------------------------------
extern "C" void kernel_launch(void* const* d_in, const int* in_sizes, int n_in,
                              void* d_out, int out_size, void* d_ws, size_t ws_size,
                              hipStream_t stream) {
    const float*     x   = (const float*)d_in[0];
    const long long* ei  = (const long long*)d_in[1];   // int64 [2,E]
    const float*     W1  = (const float*)d_in[2];
    const float*     b1  = (const float*)d_in[3];
    const float*     W2  = (const float*)d_in[4];
    const float*     b2  = (const float*)d_in[5];
    const float*     Wf1 = (const float*)d_in[6];
    const float*     bf1 = (const float*)d_in[7];
    const float*     Wf2 = (const float*)d_in[8];
    const float*     bf2 = (const float*)d_in[9];

    const int       N = in_sizes[0] / 128;
    const long long E = (long long)in_sizes[1] / 2;

    // Workspace layout (floats): dinv | h1 | agg1 | h2 | agg2  (73*N floats)
    float* ws   = (float*)d_ws;
    float* dinv = ws;                 ws += N;
    float* h1   = ws;                 ws += (size_t)N * 12;
    float* agg1 = ws;                 ws += (size_t)N * 12;
    float* h2   = ws;                 ws += (size_t)N * 24;
    float* agg2 = ws;                 /* ws += (size_t)N * 24; */

    const int B   = 256;
    const int nbN = (N + B - 1) / B;
    const int nbE = (int)((E + B - 1) / B);
    const int ntiles = (N + 15) / 16;
    const int wpb    = B / 32;                         // 8 waves per block
    const int nbT    = (ntiles + wpb - 1) / wpb;

    // degree -> dinv
    k_deg_init <<<nbN, B, 0, stream>>>(dinv, N);
    k_deg_count<<<nbE, B, 0, stream>>>(ei, dinv, E);
    k_deg_rsqrt<<<nbN, B, 0, stream>>>(dinv, N);

    // layer 1
    k_gemm1_wmma<<<nbT, B, 0, stream>>>(x, W1, h1, ntiles, N);
    k_selfloop<12><<<nbN, B, 0, stream>>>(h1, dinv, agg1, N);
    k_scatter<12><<<nbE, B, 0, stream>>>(ei, h1, dinv, agg1, E);

    // layer 2 (ReLU+bias of layer 1 fused into the GEMM2 A-load)
    k_gemm2_wmma<<<nbT, B, 0, stream>>>(agg1, b1, W2, h2, ntiles, N);
    k_selfloop<24><<<nbN, B, 0, stream>>>(h2, dinv, agg2, N);
    k_scatter<24><<<nbE, B, 0, stream>>>(ei, h2, dinv, agg2, E);

    // fused MLP head -> logits
    k_mlp<<<nbN, B, 0, stream>>>(agg2, b2, Wf1, bf1, Wf2, bf2, (float*)d_out, N);
}